// NaiveRegressionOutput_9758165697298
// MI455X (gfx1250) — compile-verified
//
#include <hip/hip_runtime.h>
#include <math.h>

// Problem constants (match reference)
#define B_DIM   128
#define N_DIM   2048
#define H_DIM   768
#define H2_DIM  1536
#define NCHUNK  8
#define NPART   (N_DIM / NCHUNK)   // 256
#define MAX_ANS 15

typedef float v2f __attribute__((ext_vector_type(2)));
typedef float v4f __attribute__((ext_vector_type(4)));
typedef float v8f __attribute__((ext_vector_type(8)));

// ---------------------------------------------------------------------------
// Kernel 1: streaming partial mean over N. 805 MB of f32 traffic -> this is
// the bandwidth-bound hot loop. float4 (b128) non-temporal coalesced loads.
// grid = (NCHUNK, B), block = 192 (one lane per float4 column of H=768).
// ---------------------------------------------------------------------------
__global__ void __launch_bounds__(192)
pool_partial_kernel(const float* __restrict__ v_emb, float* __restrict__ part) {
    const int h4 = threadIdx.x;       // 0..191  (float4 column)
    const int c  = blockIdx.x;        // N-chunk
    const int b  = blockIdx.y;        // batch row

    const v4f* __restrict__ src =
        (const v4f*)v_emb + (size_t)(b * N_DIM + c * NPART) * (H_DIM / 4) + h4;

    v4f acc = {0.0f, 0.0f, 0.0f, 0.0f};
#pragma unroll 8
    for (int n = 0; n < NPART; ++n) {
        v4f x = __builtin_nontemporal_load(src);   // TH=NT: stream, don't pollute L2
        acc += x;
        src += (H_DIM / 4);
    }

    v4f* __restrict__ dst =
        (v4f*)part + (size_t)(c * B_DIM + b) * (H_DIM / 4) + h4;
    *dst = acc;
}

// ---------------------------------------------------------------------------
// Kernel 2: reduce NCHUNK partials -> pooled[B][H], scale by 1/N.
// B*H/4 = 24576 float4 lanes; grid = 96, block = 256.
// ---------------------------------------------------------------------------
__global__ void __launch_bounds__(256)
pool_reduce_kernel(const v4f* __restrict__ part, v4f* __restrict__ pooled) {
    const int idx = blockIdx.x * 256 + threadIdx.x;    // 0 .. B*H/4-1
    v4f acc = part[idx];
#pragma unroll
    for (int c = 1; c < NCHUNK; ++c)
        acc += part[(size_t)c * (B_DIM * H_DIM / 4) + idx];
    pooled[idx] = acc * (1.0f / (float)N_DIM);
}

// ---------------------------------------------------------------------------
// Kernel 3: h = GELU(pooled @ W1 + b1) via V_WMMA_F32_16X16X4_F32.
// One wave owns one 16x16 tile of h[128,1536]; K looped 768/4 = 192 WMMAs.
// Fragment layouts per CDNA5 ISA 7.12.2 (32-bit A 16x4 / C-D 16x16):
//   A: lane<16 -> M=lane,  K pair {0,1}; lane>=16 -> M=lane-16, K pair {2,3}
//   B: lane<16 -> N=lane,  K pair {0,1}; lane>=16 -> N=lane-16, K pair {2,3}
//   C/D: VGPR r holds M = r (lanes 0-15) / M = r+8 (lanes 16-31), N = lane&15
// grid = (24, 8)  [4 N-tiles per block x 8 M-tiles], block = 128 (4 waves).
// ---------------------------------------------------------------------------
__global__ void __launch_bounds__(128)
mlp_wmma_kernel(const float* __restrict__ pooled, const float* __restrict__ W1,
                const float* __restrict__ b1, float* __restrict__ hbuf) {
    const int lane = threadIdx.x & 31;
    const int wave = threadIdx.x >> 5;
    const int n0   = (blockIdx.x * 4 + wave) * 16;   // output column tile
    const int m0   = blockIdx.y * 16;                // output row tile
    const int col  = lane & 15;
    const int hi   = lane >> 4;                      // 0: lanes 0-15, 1: lanes 16-31
    const int koff = hi << 1;                        // K sub-offset {0,2}

    v8f acc = {};

    const float* __restrict__ aptr = pooled + (size_t)(m0 + col) * H_DIM + koff;
    const float* __restrict__ bptr = W1 + (size_t)koff * H2_DIM + n0 + col;

    for (int k = 0; k < H_DIM; k += 4) {
        v2f a = *(const v2f*)(aptr + k);             // pooled[m0+col][k+koff .. +1]
        v2f bb;
        bb.x = bptr[(size_t)k * H2_DIM];             // W1[k+koff  ][n0+col]
        bb.y = bptr[(size_t)(k + 1) * H2_DIM];       // W1[k+koff+1][n0+col]
        acc = __builtin_amdgcn_wmma_f32_16x16x4_f32(
            /*neg_a=*/false, a, /*neg_b=*/false, bb,
            /*c_mod=*/(short)0, acc, /*reuse_a=*/false, /*reuse_b=*/false);
    }

    const float bias = b1[n0 + col];
#pragma unroll
    for (int r = 0; r < 8; ++r) {
        const int m = m0 + (hi ? (8 + r) : r);
        float x = acc[r] + bias;
        // exact GELU: 0.5*x*(1+erf(x/sqrt(2)))
        float g = 0.5f * x * (1.0f + erff(x * 0.70710678118654752f));
        hbuf[(size_t)m * H2_DIM + n0 + col] = g;
    }
}

// ---------------------------------------------------------------------------
// Kernel 4: per-row LayerNorm over 2H=1536, dot with W2, pred + one-hot.
// grid = B, block = 256 (each thread owns 6 columns).
// ---------------------------------------------------------------------------
__global__ void __launch_bounds__(256)
ln_head_kernel(const float* __restrict__ hbuf, const float* __restrict__ gamma,
               const float* __restrict__ beta, const float* __restrict__ W2,
               const float* __restrict__ b2, float* __restrict__ out) {
    __shared__ float sA[256];
    __shared__ float sB[256];
    __shared__ float sbc[2];
    __shared__ int   said;

    const int b = blockIdx.x;
    const int t = threadIdx.x;

    float v[6];
    float s = 0.0f, q = 0.0f;
#pragma unroll
    for (int i = 0; i < 6; ++i) {
        v[i] = hbuf[(size_t)b * H2_DIM + i * 256 + t];
        s += v[i];
        q += v[i] * v[i];
    }
    sA[t] = s; sB[t] = q;
    __syncthreads();
    for (int off = 128; off > 0; off >>= 1) {
        if (t < off) { sA[t] += sA[t + off]; sB[t] += sB[t + off]; }
        __syncthreads();
    }
    if (t == 0) {
        float mu  = sA[0] * (1.0f / (float)H2_DIM);
        float var = sB[0] * (1.0f / (float)H2_DIM) - mu * mu;
        sbc[0] = mu;
        sbc[1] = rsqrtf(var + 1e-5f);
    }
    __syncthreads();
    const float mu = sbc[0], rv = sbc[1];

    float dot = 0.0f;
#pragma unroll
    for (int i = 0; i < 6; ++i) {
        const int idx = i * 256 + t;
        float hn = (v[i] - mu) * rv * gamma[idx] + beta[idx];
        dot += hn * W2[idx];
    }
    sA[t] = dot;
    __syncthreads();
    for (int off = 128; off > 0; off >>= 1) {
        if (t < off) sA[t] += sA[t + off];
        __syncthreads();
    }
    if (t == 0) {
        float pred = sA[0] + b2[0];
        out[b] = pred;                                  // pred output [B,1]
        float r = rintf(pred);                          // round-half-even (jnp.round)
        r = fminf(fmaxf(r, 0.0f), (float)MAX_ANS);
        said = (int)r;
    }
    __syncthreads();
    if (t < 16)
        out[B_DIM + b * 16 + t] = (t == said) ? 1.0f : 0.0f;   // one-hot [B,16]
}

// ---------------------------------------------------------------------------
extern "C" void kernel_launch(void* const* d_in, const int* in_sizes, int n_in,
                              void* d_out, int out_size, void* d_ws, size_t ws_size,
                              hipStream_t stream) {
    const float* v_emb = (const float*)d_in[0];   // [B,N,H]
    const float* W1    = (const float*)d_in[1];   // [H,2H]
    const float* b1    = (const float*)d_in[2];   // [2H]
    const float* gamma = (const float*)d_in[3];   // [2H]
    const float* beta  = (const float*)d_in[4];   // [2H]
    const float* W2    = (const float*)d_in[5];   // [2H,1]
    const float* b2    = (const float*)d_in[6];   // [1]
    float* out = (float*)d_out;                   // [B] pred ++ [B,16] logits

    float* ws     = (float*)d_ws;
    float* part   = ws;                                        // NCHUNK*B*H
    float* pooled = part + (size_t)NCHUNK * B_DIM * H_DIM;     // B*H
    float* hbuf   = pooled + (size_t)B_DIM * H_DIM;            // B*2H

    pool_partial_kernel<<<dim3(NCHUNK, B_DIM), 192, 0, stream>>>(v_emb, part);
    pool_reduce_kernel<<<dim3((B_DIM * H_DIM / 4) / 256), 256, 0, stream>>>(
        (const v4f*)part, (v4f*)pooled);
    mlp_wmma_kernel<<<dim3(24, 8), 128, 0, stream>>>(pooled, W1, b1, hbuf);
    ln_head_kernel<<<dim3(B_DIM), 256, 0, stream>>>(hbuf, gamma, beta, W2, b2, out);
}